// SplitMind_60687887892956
// MI455X (gfx1250) — compile-verified
//
#include <hip/hip_runtime.h>
#include <hip/hip_bf16.h>

typedef float v2f __attribute__((ext_vector_type(2)));
typedef float v8f __attribute__((ext_vector_type(8)));

#define D_DIM 512
#define N_DIM 32768
#define E_DIM 8
#define LDS_STRIDE 516  // 512 + 4 dword pad -> 16 rows land on distinct LDS banks

// ---------------------------------------------------------------------------
// Kernel A: Wsum[f,d] = sum_e W[e,f,d];  bsum[f] = sum_e b[e,f]
// float4-vectorized: 65536 float4 elements of Wsum, 128 float4 of bsum.
// ---------------------------------------------------------------------------
__global__ __launch_bounds__(256) void SplitMind_reduce_experts(
    const float* __restrict__ W, const float* __restrict__ b,
    float* __restrict__ Wsum, float* __restrict__ bsum) {
  const int i = blockIdx.x * 256 + threadIdx.x;  // float4 index, grid covers 65536
  const float4* W4 = (const float4*)W;
  float4 acc = W4[i];
#pragma unroll
  for (int e = 1; e < E_DIM; ++e) {
    float4 v = W4[(size_t)e * (D_DIM * D_DIM / 4) + i];
    acc.x += v.x; acc.y += v.y; acc.z += v.z; acc.w += v.w;
  }
  ((float4*)Wsum)[i] = acc;

  if (i < D_DIM / 4) {
    const float4* b4 = (const float4*)b;
    float4 ab = b4[i];
#pragma unroll
    for (int e = 1; e < E_DIM; ++e) {
      float4 v = b4[e * (D_DIM / 4) + i];
      ab.x += v.x; ab.y += v.y; ab.z += v.z; ab.w += v.w;
    }
    ((float4*)bsum)[i] = ab;
  }
}

// ---------------------------------------------------------------------------
// Kernel B: out[n,f] = w[n] * (sum_d X[n,d]*Wsum[f,d] + bsum[f])
// Block = 8 waves (256 thr). Block owns 16 rows of X (staged in LDS once).
// Wave w owns columns [64w, 64w+64): 4 x (16x16) f32 accum tiles.
// K-loop: 128 steps of V_WMMA_F32_16X16X4_F32 per tile.
// ---------------------------------------------------------------------------
__global__ __launch_bounds__(256) void SplitMind_moe_gemm(
    const float* __restrict__ X, const float* __restrict__ wgt,
    const float* __restrict__ Wsum, const float* __restrict__ bsum,
    float* __restrict__ out) {
  __shared__ float xs[16 * LDS_STRIDE];

  const int r0   = blockIdx.x * 16;
  const int tid  = threadIdx.x;
  const int lane = tid & 31;
  const int wave = tid >> 5;

  // --- Stage X[r0:r0+16, 0:512] -> LDS (16 rows, padded stride) ---
  // 2048 float4 elements total; 8 per thread.
#pragma unroll
  for (int it = 0; it < 8; ++it) {
    int i   = tid + it * 256;      // float4 index
    int row = i >> 7;              // 128 float4 per row
    int c4  = i & 127;
    float4 v = ((const float4*)(X + (size_t)(r0 + row) * D_DIM))[c4];
    float* dst = &xs[row * LDS_STRIDE + c4 * 4];
    dst[0] = v.x; dst[1] = v.y; dst[2] = v.z; dst[3] = v.w;
  }
  __syncthreads();

  // --- WMMA fragment addressing (ISA 7.12.2 layouts) ---
  const int m    = lane & 15;          // row within A tile / col within B,C,D tile
  const int hi   = lane >> 4;          // 0: K=k,k+1   1: K=k+2,k+3
  const int koff = 2 * hi;
  const int fb   = wave * 64;

  // A-fragment source: X row m, columns k+koff .. k+koff+1 (8-byte LDS load)
  const float* aptr = &xs[m * LDS_STRIDE + koff];
  // B-fragment sources: B[k,f] = Wsum[f,k]; lane loads Wsum[f, k+koff .. +1]
  const float* bptr0 = Wsum + (size_t)(fb +  0 + m) * D_DIM + koff;
  const float* bptr1 = Wsum + (size_t)(fb + 16 + m) * D_DIM + koff;
  const float* bptr2 = Wsum + (size_t)(fb + 32 + m) * D_DIM + koff;
  const float* bptr3 = Wsum + (size_t)(fb + 48 + m) * D_DIM + koff;

  v8f acc0 = {}, acc1 = {}, acc2 = {}, acc3 = {};

#pragma unroll 4
  for (int k = 0; k < D_DIM; k += 4) {
    v2f a  = *(const v2f*)(aptr  + k);
    v2f b0 = *(const v2f*)(bptr0 + k);
    v2f b1 = *(const v2f*)(bptr1 + k);
    v2f b2 = *(const v2f*)(bptr2 + k);
    v2f b3 = *(const v2f*)(bptr3 + k);
    acc0 = __builtin_amdgcn_wmma_f32_16x16x4_f32(false, a, false, b0, (short)0, acc0, false, false);
    acc1 = __builtin_amdgcn_wmma_f32_16x16x4_f32(false, a, false, b1, (short)0, acc1, false, false);
    acc2 = __builtin_amdgcn_wmma_f32_16x16x4_f32(false, a, false, b2, (short)0, acc2, false, false);
    acc3 = __builtin_amdgcn_wmma_f32_16x16x4_f32(false, a, false, b3, (short)0, acc3, false, false);
  }

  // --- Epilogue: out[row,col] = (acc + bsum[col]) * w[row] ---
  // C/D layout: VGPR j holds M=j (lanes 0-15) / M=j+8 (lanes 16-31), N=lane&15.
  v8f accs[4] = {acc0, acc1, acc2, acc3};
  float bvals[4];
#pragma unroll
  for (int t = 0; t < 4; ++t) bvals[t] = bsum[fb + t * 16 + m];

#pragma unroll
  for (int j = 0; j < 8; ++j) {
    const int row = r0 + j + 8 * hi;
    const float wv = wgt[row];
    float* orow = out + (size_t)row * D_DIM + fb + m;
#pragma unroll
    for (int t = 0; t < 4; ++t) {
      orow[t * 16] = (accs[t][j] + bvals[t]) * wv;
    }
  }
}

extern "C" void kernel_launch(void* const* d_in, const int* in_sizes, int n_in,
                              void* d_out, int out_size, void* d_ws, size_t ws_size,
                              hipStream_t stream) {
  const float* X   = (const float*)d_in[0];  // [N, D]
  const float* wgt = (const float*)d_in[1];  // [N, 1]
  const float* W   = (const float*)d_in[2];  // [E, D, D]
  const float* b   = (const float*)d_in[3];  // [E, D]
  float* out  = (float*)d_out;               // [N, D]
  float* Wsum = (float*)d_ws;                // D*D floats
  float* bsum = Wsum + D_DIM * D_DIM;        // D floats

  // Kernel A: 65536 float4 elements / 256 threads = 256 blocks
  SplitMind_reduce_experts<<<256, 256, 0, stream>>>(W, b, Wsum, bsum);
  // Kernel B: one block per 16-row strip of X
  SplitMind_moe_gemm<<<N_DIM / 16, 256, 0, stream>>>(X, wgt, Wsum, bsum, out);
}